// NSLayer_69380901700219
// MI455X (gfx1250) — compile-verified
//
#include <hip/hip_runtime.h>

typedef float v2f __attribute__((ext_vector_type(2)));
typedef float v4f __attribute__((ext_vector_type(4)));
typedef float v8f __attribute__((ext_vector_type(8)));

#define WAVES_PER_BLOCK 8
#define TILES_PER_WAVE 4
// per-wave LDS: 128 words x-staging + 256 words D-spill
#define LDS_WORDS_PER_WAVE 384

__device__ __forceinline__ void lds_fence() {
    // DS ops from one wave are processed in order; this waits for outstanding
    // ds_stores so subsequent cross-lane ds_loads observe them, and the
    // memory clobber stops the compiler from hoisting loads above stores.
    asm volatile("s_wait_dscnt 0" ::: "memory");
}

__device__ __forceinline__ v8f wmma_f32_k4(v2f a, v2f b, v8f c) {
    // D = A(16x4) * B(4x16) + C, fp32
    return __builtin_amdgcn_wmma_f32_16x16x4_f32(
        false, a, false, b, (short)0, c, false, false);
}

__global__ __launch_bounds__(256)
void NSLayer_poly_kernel(const float* __restrict__ x,
                         const float* __restrict__ wt,
                         float* __restrict__ out,
                         int ntiles) {
    __shared__ float lds[WAVES_PER_BLOCK * LDS_WORDS_PER_WAVE];

    const int tid  = threadIdx.x;
    const int wid  = tid >> 5;
    const int lane = tid & 31;
    const int gw   = blockIdx.x * WAVES_PER_BLOCK + wid;  // global wave id

    float* stage = &lds[wid * LDS_WORDS_PER_WAVE]; // 128 words: x for 2 matrices
    float* spill = stage + 128;                    // 256 words: 16x16 D tile

    // weights (uniform across wave -> one scalar b256 load)
    const float w0 = wt[0], w1 = wt[1], w2 = wt[2], w3 = wt[3];
    const float w4 = wt[4], w5 = wt[5], w6 = wt[6], w7 = wt[7];
    const float wseq[6] = { w5, w4, w3, w2, w1, w0 };
    const float s7 = 1.0f + w7;

    // ---- per-lane constants (loop-invariant: computed once per wave) ----
    const int half = lane >> 4;   // which 16-lane half
    const int lm   = lane & 15;   // 0..15
    const int li   = lm >> 3;     // matrix (0/1) for lane-major layouts
    const int lr   = lm & 7;      // row within matrix

    // A-operand gather of x (== B-operand gather of x^T): base word, b64 pairs
    const int wL = li * 64 + lr * 8 + 2 * half;
    // B-operand gather of [x1,x2] from staging: word (li*64 + k*8 + lr)
    int xbw[4];
    // A-operand gather of XX from spill: word (k + 24*li)*8 + lr, plus identity
    int aaw[4];
    float idA[4];
#pragma unroll
    for (int j = 0; j < 2; ++j)
#pragma unroll
        for (int q = 0; q < 2; ++q) {
            const int k = 4 * j + q + 2 * half;
            xbw[j * 2 + q] = li * 64 + k * 8 + lr;
            aaw[j * 2 + q] = (k + 24 * li) * 8 + lr;
            idA[j * 2 + q] = (lr == k) ? 1.0f : 0.0f;
        }
    // B-operand regather of D from spill (q contiguous -> b64 pairs)
    const int wS = (lm + 16 * li) * 8 + 2 * half;
    // output gather: this lane writes flat words f = lane*4 + j of the tile
    const int omat = half;              // (lane*4)/64
    const int om   = (lane >> 1) & 7;   // row within matrix
    const int on0  = (lane & 1) * 4;    // starting col (0 or 4)
    int ow[4];
#pragma unroll
    for (int j = 0; j < 4; ++j)
        ow[j] = (on0 + j + 24 * omat) * 8 + om;

#pragma unroll
    for (int it = 0; it < TILES_PER_WAVE; ++it) {
        const int tile = gw * TILES_PER_WAVE + it;
        if (tile >= ntiles) break;   // wave-uniform

        // ---- load 2 matrices (128 floats) coalesced, stage to LDS ----
        const size_t base = (size_t)tile * 128;
        v4f xr = *(const v4f*)(x + base + (size_t)lane * 4);
        *(v4f*)(stage + lane * 4) = xr;
        lds_fence();

        // ---- gather x in A-operand layout of [x1;x2] (b64 per slice) ----
        v2f ax0 = *(const v2f*)(stage + wL);      // slice j=0
        v2f ax1 = *(const v2f*)(stage + wL + 4);  // slice j=1

        // ---- gather x in B-operand layout of [x1, x2] ----
        float xb[4];
#pragma unroll
        for (int e = 0; e < 4; ++e) xb[e] = stage[xbw[e]];

        // ---- XX = diag(x1 x1^T, x2 x2^T): A/B register images coincide ----
        v8f d = {};
        d = wmma_f32_k4(ax0, ax0, d);
        d = wmma_f32_k4(ax1, ax1, d);

        // ---- spill XX, build Astack = I - XX in A-operand layout ----
        {
            v4f lo = { d[0], d[1], d[2], d[3] };
            v4f hi = { d[4], d[5], d[6], d[7] };
            *(v4f*)(spill + lane * 8)     = lo;
            *(v4f*)(spill + lane * 8 + 4) = hi;
        }
        lds_fence();
        float aA[4];
#pragma unroll
        for (int e = 0; e < 4; ++e) aA[e] = idA[e] - spill[aaw[e]];
        v2f A0 = { aA[0], aA[1] };
        v2f A1 = { aA[2], aA[3] };

        // ---- Horner chain: t = w6*A*x + w5*x; then t = A*t + w_i*x ----
        v2f tB0 = { w6 * xb[0], w6 * xb[1] };
        v2f tB1 = { w6 * xb[2], w6 * xb[3] };

#pragma unroll
        for (int s = 0; s < 6; ++s) {
            v8f acc = {};
            acc = wmma_f32_k4(A0, tB0, acc);
            acc = wmma_f32_k4(A1, tB1, acc);
            v4f lo = { acc[0], acc[1], acc[2], acc[3] };
            v4f hi = { acc[4], acc[5], acc[6], acc[7] };
            *(v4f*)(spill + lane * 8)     = lo;
            *(v4f*)(spill + lane * 8 + 4) = hi;
            lds_fence();
            v2f c0 = *(const v2f*)(spill + wS);
            v2f c1 = *(const v2f*)(spill + wS + 4);
            const float wi = wseq[s];
            tB0 = v2f{ c0[0] + wi * xb[0], c0[1] + wi * xb[1] };
            tB1 = v2f{ c1[0] + wi * xb[2], c1[1] + wi * xb[3] };
        }

        // ---- final: out = x + (A*t + w7*x) = (1+w7)*x + A*t ----
        v8f acc = {};
        acc = wmma_f32_k4(A0, tB0, acc);
        acc = wmma_f32_k4(A1, tB1, acc);
        {
            v4f lo = { acc[0], acc[1], acc[2], acc[3] };
            v4f hi = { acc[4], acc[5], acc[6], acc[7] };
            *(v4f*)(spill + lane * 8)     = lo;
            *(v4f*)(spill + lane * 8 + 4) = hi;
        }
        lds_fence();

        v4f o;
#pragma unroll
        for (int j = 0; j < 4; ++j)
            o[j] = s7 * xr[j] + spill[ow[j]];
        *(v4f*)(out + base + (size_t)lane * 4) = o;
    }
}

extern "C" void kernel_launch(void* const* d_in, const int* in_sizes, int n_in,
                              void* d_out, int out_size, void* d_ws, size_t ws_size,
                              hipStream_t stream) {
    const float* x  = (const float*)d_in[0];
    const float* wt = (const float*)d_in[1];
    float* out = (float*)d_out;

    const int nmat   = in_sizes[0] / 64;  // 2048*128 = 262144 matrices of 8x8
    const int ntiles = nmat / 2;          // 2 matrices per 16x16 WMMA tile
    const int tiles_per_block = WAVES_PER_BLOCK * TILES_PER_WAVE;
    const int blocks = (ntiles + tiles_per_block - 1) / tiles_per_block;

    NSLayer_poly_kernel<<<blocks, WAVES_PER_BLOCK * 32, 0, stream>>>(x, wt, out, ntiles);
}